// SurfEval_30846455119883
// MI455X (gfx1250) — compile-verified
//
#include <hip/hip_runtime.h>
#include <stdint.h>

// Problem constants (from reference): B=8, M=N=128, DIM=3, P=Q=3, OUT=512
#define BATCH     8
#define MCTRL     128
#define NCTRL     128
#define OUTS      512
#define TPB       1024              // 32 wave32 waves per block
#define PPT       8                 // pixels per thread
#define TILE_PIX  (TPB * PPT)       // 8192 pixels per block
#define SLICE_F4  (MCTRL * NCTRL)   // 16384 float4 = 256 KB (fits 320 KB LDS)
#define NDMA      4                 // split DMA across 4 waves (4 x 64 KB)
#define DMA_BYTES (SLICE_F4 * 16 / NDMA)   // 65536 bytes per descriptor
#define DMA_ELEM8 (DMA_BYTES / 8)          // 8192 8-byte elements

typedef unsigned int u32x4 __attribute__((ext_vector_type(4)));
typedef int          i32x8 __attribute__((ext_vector_type(8)));
typedef int          i32x4 __attribute__((ext_vector_type(4)));

#if defined(__has_builtin)
#if __has_builtin(__builtin_amdgcn_tensor_load_to_lds)
#define HAVE_TDM 1
#endif
#endif

__global__ void __launch_bounds__(TPB)
surfeval_tdm_kernel(const float* __restrict__ ctrl,   // (8,128,128,4) f32
                    const float* __restrict__ Nu,     // (512,512,4)   f32
                    const float* __restrict__ Nv,     // (512,512,4)   f32
                    const int*   __restrict__ uspan,  // (512,512)     i32
                    const int*   __restrict__ vspan,  // (512,512)     i32
                    float*       __restrict__ out)    // (8,512,512,3) f32
{
    extern __shared__ float4 smem[];  // 16384 float4 = 256 KB: one batch slice
    const int b = blockIdx.y;
    const int tileStart = blockIdx.x * TILE_PIX;
    const uint64_t sliceBase =
        (uint64_t)(uintptr_t)(ctrl + (size_t)b * SLICE_F4 * 4);

#if defined(HAVE_TDM)
    // ---- TDM: DMA the 256 KB ctrl_pts[b] slice into LDS as 4 concurrent ----
    // ---- 64 KB descriptors, one per wave (TDM is unordered across waves) ----
    const unsigned wid = threadIdx.x >> 5;           // wave id (wave32)
    if (wid < NDMA) {
        const unsigned ldsOff = wid * DMA_BYTES;
        const uint64_t gaddr  = sliceBase + (uint64_t)ldsOff;
        // D# group 0 (ISA 8.3): count=1 | lds_addr | global_addr | type=2
        u32x4 g0;
        g0[0] = 1u;                                        // count=1 (valid)
        g0[1] = ldsOff;                                    // lds_addr (bytes)
        g0[2] = (unsigned)(gaddr & 0xFFFFFFFFu);           // global_addr[31:0]
        g0[3] = (unsigned)((gaddr >> 32) & 0x1FFFFFFu)     // global_addr[56:32]
              | (2u << 30);                                // type = 2 ("image")
        // D# group 1 (ISA 8.4): 1-D copy, data_size=8B, 8192 elements
        i32x8 g1;
        g1[0] = (int)(3u << 16);                 // data_size=3 (8 B)
        g1[1] = (int)((unsigned)DMA_ELEM8 << 16);// tensor_dim0[15:0] @bits63:48
        g1[2] = 0;                               // tensor_dim0 hi, tensor_dim1
        g1[3] = (int)((unsigned)DMA_ELEM8 << 16);// tile_dim0 @bits127:112
        g1[4] = 0;                               // tile_dim1=0, tile_dim2=0
        g1[5] = DMA_ELEM8;                       // tensor_dim0_stride[31:0]
        g1[6] = 0;
        g1[7] = 0;
        i32x4 g2 = {0, 0, 0, 0};
        i32x4 g3 = {0, 0, 0, 0};
#if __has_include(<hip/amd_detail/amd_gfx1250_TDM.h>)
        i32x8 g4 = {0, 0, 0, 0, 0, 0, 0, 0};    // 6-arg therock toolchain form
        __builtin_amdgcn_tensor_load_to_lds(g0, g1, g2, g3, g4, 0);
#else
        __builtin_amdgcn_tensor_load_to_lds(g0, g1, g2, g3, 0); // ROCm 7.2 form
#endif
        __builtin_amdgcn_s_wait_tensorcnt(0);   // each issuing wave drains own cnt
    }
#else
    // ---- Fallback: cooperative float4 copy (coalesced) ----
    {
        const float4* src = (const float4*)(uintptr_t)sliceBase;
        for (int i = threadIdx.x; i < SLICE_F4; i += TPB) smem[i] = src[i];
    }
#endif

    // ---- While the DMA is in flight: prefetch this block's pixel stream ----
    // Nu/Nv tiles: 8192 px * 16 B = 128 KB = 1024 x 128 B lines (1 per thread)
    // span tiles :  8192 px *  4 B =  32 KB =  256 x 128 B lines
    {
        const char* nuB = (const char*)(Nu + (size_t)tileStart * 4);
        const char* nvB = (const char*)(Nv + (size_t)tileStart * 4);
        __builtin_prefetch(nuB + (size_t)threadIdx.x * 128, 0, 3);
        __builtin_prefetch(nvB + (size_t)threadIdx.x * 128, 0, 3);
        if (threadIdx.x < 256u) {
            const char* usB = (const char*)(uspan + tileStart);
            const char* vsB = (const char*)(vspan + tileStart);
            __builtin_prefetch(usB + (size_t)threadIdx.x * 128, 0, 3);
            __builtin_prefetch(vsB + (size_t)threadIdx.x * 128, 0, 3);
        }
    }

    __syncthreads();   // join the 4 DMA waves; LDS visible to whole workgroup

    // ---- Gather + bilinear weighted sum from LDS ----
#pragma unroll
    for (int k = 0; k < PPT; ++k) {
        const int pix = tileStart + k * TPB + (int)threadIdx.x;  // coalesced
        const float4 nu = *(const float4*)(Nu + (size_t)pix * 4);
        const float4 nv = *(const float4*)(Nv + (size_t)pix * 4);
        const int iu0 = uspan[pix] - 3;   // uspan in [3,127] -> rows iu0..iu0+3
        const int iv0 = vspan[pix] - 3;

        const float nuk[4] = {nu.x, nu.y, nu.z, nu.w};
        const float nvk[4] = {nv.x, nv.y, nv.z, nv.w};

        float ax = 0.f, ay = 0.f, az = 0.f, aw = 0.f;
        const float4* rowBase = smem + iu0 * NCTRL + iv0;
#pragma unroll
        for (int l = 0; l < 4; ++l) {
            const float4* r4 = rowBase + l * NCTRL;   // 4 contiguous float4
            float sx = 0.f, sy = 0.f, sz = 0.f, sw = 0.f;
#pragma unroll
            for (int r = 0; r < 4; ++r) {
                const float4 p = r4[r];               // ds_load_b128 (16B aligned)
                const float w = nvk[r];
                sx = fmaf(w, p.x, sx);
                sy = fmaf(w, p.y, sy);
                sz = fmaf(w, p.z, sz);
                sw = fmaf(w, p.w, sw);
            }
            const float wl = nuk[l];
            ax = fmaf(wl, sx, ax);
            ay = fmaf(wl, sy, ay);
            az = fmaf(wl, sz, az);
            aw = fmaf(wl, sw, aw);
        }

        const float inv = 1.0f / aw;                  // perspective divide
        const size_t o = ((size_t)b * OUTS * OUTS + (size_t)pix) * 3;
        out[o + 0] = ax * inv;
        out[o + 1] = ay * inv;
        out[o + 2] = az * inv;
    }
}

extern "C" void kernel_launch(void* const* d_in, const int* in_sizes, int n_in,
                              void* d_out, int out_size, void* d_ws, size_t ws_size,
                              hipStream_t stream) {
    (void)in_sizes; (void)n_in; (void)out_size; (void)d_ws; (void)ws_size;
    const float* ctrl = (const float*)d_in[0];   // ctrl_pts
    const float* Nu   = (const float*)d_in[1];   // Nu_uv
    const float* Nv   = (const float*)d_in[2];   // Nv_uv
    const int*   us   = (const int*)d_in[3];     // uspan_uv
    const int*   vs   = (const int*)d_in[4];     // vspan_uv
    float*       out  = (float*)d_out;           // (8,512,512,3)

    dim3 grid(OUTS * OUTS / TILE_PIX, BATCH, 1); // (32, 8)
    dim3 block(TPB, 1, 1);
    size_t shmem = (size_t)SLICE_F4 * sizeof(float4);  // 262144 bytes
    surfeval_tdm_kernel<<<grid, block, shmem, stream>>>(ctrl, Nu, Nv, us, vs, out);
}